// GaugeAligner_6880537608279
// MI455X (gfx1250) — compile-verified
//
#include <hip/hip_runtime.h>

// ---------------------------------------------------------------------------
// GaugeAligner loss for MI455X (gfx1250, wave32).
// Memory-bound (~147 MB @ 23.3 TB/s ~ 6.3 us). One thread per lattice site,
// float4 (b128) loads for U matrices, recompute g at forward neighbors from
// theta (L2-resident). Reduction uses V_WMMA_F32_16X16X4_F32 with a ones
// B-matrix as a one-shot 32-lane adder.
// ---------------------------------------------------------------------------

#define LAT 1024
#define LOG2L 10
#define LMASK (LAT - 1)
#define SU2_EPS 1e-8f

typedef float v2f __attribute__((ext_vector_type(2)));
typedef float v8f __attribute__((ext_vector_type(8)));

struct c32 { float re, im; };

__device__ __forceinline__ c32 cmul(c32 a, c32 b) {
    return { a.re * b.re - a.im * b.im, a.re * b.im + a.im * b.re };
}
// a * conj(b)
__device__ __forceinline__ c32 cmulj(c32 a, c32 b) {
    return { a.re * b.re + a.im * b.im, a.im * b.re - a.re * b.im };
}
__device__ __forceinline__ c32 cadd(c32 a, c32 b) {
    return { a.re + b.re, a.im + b.im };
}

// g = exp(i theta.sigma) = cos(n) I + i sin(n)/n (theta.sigma)
//   g00 = c + i s t3,  g01 =  s t2 + i s t1
//   g10 = -s t2 + i s t1,  g11 = c - i s t3
__device__ __forceinline__ void make_g(const float* __restrict__ t, c32 g[2][2]) {
    float t1 = t[0], t2 = t[1], t3 = t[2];
    float n = sqrtf(t1 * t1 + t2 * t2 + t3 * t3);
    float s, c;
    __sincosf(n, &s, &c);
    float sinc = (n > SU2_EPS) ? (s / n) : 1.0f;
    float a = sinc * t1, b = sinc * t2, d = sinc * t3;
    g[0][0] = {  c,  d };
    g[0][1] = {  b,  a };
    g[1][0] = { -b,  a };
    g[1][1] = {  c, -d };
}

// Layout per (site, mu): [a][b][re/im] -> two float4s.
__device__ __forceinline__ void load_mat(const float4* __restrict__ p, c32 U[2][2]) {
    float4 q0 = p[0], q1 = p[1];
    U[0][0] = { q0.x, q0.y };  U[0][1] = { q0.z, q0.w };
    U[1][0] = { q1.x, q1.y };  U[1][1] = { q1.z, q1.w };
}

// sum over entries |P - g * U * h^dagger|^2
__device__ __forceinline__ float site_term(const c32 g[2][2], const c32 U[2][2],
                                           const c32 h[2][2], const c32 P[2][2]) {
    c32 t[2][2];
#pragma unroll
    for (int a = 0; a < 2; ++a)
#pragma unroll
        for (int c = 0; c < 2; ++c)
            t[a][c] = cadd(cmul(g[a][0], U[0][c]), cmul(g[a][1], U[1][c]));
    float s = 0.0f;
#pragma unroll
    for (int a = 0; a < 2; ++a)
#pragma unroll
        for (int d = 0; d < 2; ++d) {
            c32 m = cadd(cmulj(t[a][0], h[d][0]), cmulj(t[a][1], h[d][1]));
            float dr = P[a][d].re - m.re;
            float di = P[a][d].im - m.im;
            s = fmaf(dr, dr, s);
            s = fmaf(di, di, s);
        }
    return s;
}

// Full-wave (32 lane) sum via one V_WMMA_F32_16X16X4_F32.
// A: VGPR0 = partial, VGPR1 = 0  => A[m][0]=p[m], A[m][2]=p[m+16], cols 1,3 = 0
// B: all ones                    => D[m][n] = p[m] + p[m+16]  (every column n)
// Lane n in 0..15 holds D[0..7][n]; lane n+16 holds D[8..15][n]. Sum the 8
// accumulator VGPRs locally, then fold halves with one shfl_xor(16).
// EXEC must be all-ones here (callers guarantee no divergence).
__device__ __forceinline__ float wave_reduce_wmma(float partial) {
    v2f a; a[0] = partial; a[1] = 0.0f;
    v2f b; b[0] = 1.0f;    b[1] = 1.0f;
    v8f c = {};
    v8f d = __builtin_amdgcn_wmma_f32_16x16x4_f32(
        /*neg_a=*/false, a, /*neg_b=*/false, b,
        /*c_mod=*/(short)0, c, /*reuse_a=*/false, /*reuse_b=*/false);
    float s = d[0] + d[1] + d[2] + d[3] + d[4] + d[5] + d[6] + d[7];
    s += __shfl_xor(s, 16);
    return s;  // identical in all lanes
}

__device__ __forceinline__ void block_reduce_store(float acc, float* __restrict__ dst) {
    __shared__ float lds[8];
    float w = wave_reduce_wmma(acc);
    int lane = threadIdx.x & 31;
    int wave = threadIdx.x >> 5;
    if (lane == 0) lds[wave] = w;
    __syncthreads();
    if (threadIdx.x == 0) {
        float t = 0.0f;
#pragma unroll
        for (int i = 0; i < 8; ++i) t += lds[i];
        *dst = t;
    }
}

__global__ void gauge_site_loss_kernel(const float* __restrict__ theta,
                                       const float* __restrict__ Upred,
                                       const float* __restrict__ Utrue,
                                       float* __restrict__ block_partials) {
    const int tid = blockIdx.x * blockDim.x + threadIdx.x;   // site id, exact cover
    const int x = tid >> LOG2L;
    const int y = tid & LMASK;
    const int xn = (x + 1) & LMASK;
    const int yn = (y + 1) & LMASK;

    c32 g0[2][2], gx[2][2], gy[2][2];
    make_g(theta + 3 * tid, g0);
    make_g(theta + 3 * ((xn << LOG2L) | y), gx);
    make_g(theta + 3 * ((x << LOG2L) | yn), gy);

    float acc = 0.0f;
    {   // mu = 0: neighbor along x
        c32 Ut[2][2], Up[2][2];
        load_mat((const float4*)(Utrue + 16 * tid), Ut);
        load_mat((const float4*)(Upred + 16 * tid), Up);
        acc += site_term(g0, Ut, gx, Up);
    }
    {   // mu = 1: neighbor along y
        c32 Ut[2][2], Up[2][2];
        load_mat((const float4*)(Utrue + 16 * tid + 8), Ut);
        load_mat((const float4*)(Upred + 16 * tid + 8), Up);
        acc += site_term(g0, Ut, gy, Up);
    }

    block_reduce_store(acc, block_partials + blockIdx.x);
}

__global__ void gauge_final_reduce_kernel(const float* __restrict__ partials,
                                          float* __restrict__ out,
                                          int n, float scale) {
    float acc = 0.0f;
    // n is a multiple of blockDim.x -> uniform loop, EXEC stays all-ones.
    for (int i = threadIdx.x; i < n; i += blockDim.x) acc += partials[i];
    __shared__ float lds[8];
    float w = wave_reduce_wmma(acc);
    int lane = threadIdx.x & 31;
    int wave = threadIdx.x >> 5;
    if (lane == 0) lds[wave] = w;
    __syncthreads();
    if (threadIdx.x == 0) {
        float t = 0.0f;
#pragma unroll
        for (int i = 0; i < 8; ++i) t += lds[i];
        out[0] = t * scale;
    }
}

extern "C" void kernel_launch(void* const* d_in, const int* in_sizes, int n_in,
                              void* d_out, int out_size, void* d_ws, size_t ws_size,
                              hipStream_t stream) {
    (void)in_sizes; (void)n_in; (void)out_size; (void)ws_size;
    const float* theta = (const float*)d_in[0];
    const float* Upred = (const float*)d_in[1];
    const float* Utrue = (const float*)d_in[2];
    float* out = (float*)d_out;
    float* partials = (float*)d_ws;

    const int nsites  = LAT * LAT;          // 1,048,576
    const int threads = 256;                // 8 waves / block on wave32
    const int nblocks = nsites / threads;   // 4096 (exact)

    gauge_site_loss_kernel<<<nblocks, threads, 0, stream>>>(theta, Upred, Utrue, partials);

    // final = sum(|diff|^2) / (4 entries * L*L * 2 mu)
    const float scale = 1.0f / (4.0f * (float)LAT * (float)LAT * 2.0f);
    gauge_final_reduce_kernel<<<1, threads, 0, stream>>>(partials, out, nblocks, scale);
}